// Points_75110388073241
// MI455X (gfx1250) — compile-verified
//
#include <hip/hip_runtime.h>
#include <math.h>

// MI455X / gfx1250, wave32.
// Reference: point-splatting. H=W=256, N=1024 (fixed by the reference file).
//
// Formulation:
//   t_j(pixel, n) = y*T[n][0][j] + x*T[n][1][j] - c[n][j],  c = loc . T
// => per 16-pixel tile: D = A(16x4) x B(4x16) with A=[y,x,1,0] rows,
//    one WMMA_F32_16X16X4_F32 each for tx and ty (identical D layouts),
//    then elementwise m = relu(1 - 0.5*(|tx|+|ty|)) and per-lane FMA
//    accumulation of color*alpha weights; cross-lane butterfly at the end.
//
// Inner-loop notes:
//  * |tx|+|ty| -> one VOP3 v_add_f32 with |src| modifiers (inline asm; the
//    SLP vectorizer otherwise picks v_pk_add_f32 + discrete v_and abs masks).
//  * relu(1-0.5s) == clamp01(1-0.5s) since s>=0, so it is one
//    v_fma_f32_e64 ... clamp (VOP3 CLAMP output modifier), removing the
//    separate v_max_num stage.
// Per 256 pixel-point pairs: 2 WMMA + ~28 VALU + 2 loads.

typedef float v2f __attribute__((ext_vector_type(2)));
typedef float v8f __attribute__((ext_vector_type(8)));

#define N_PTS    1024
#define H_CANVAS 256
#define W_CANVAS 256
#define N_CHUNKS (N_PTS / 16)   // 64 chunks of 16 points

__device__ __forceinline__ float abs_add(float a, float b) {
    float s;
    // VOP3 encoding carries abs source modifiers; one VALU op total.
    asm("v_add_f32_e64 %0, |%1|, |%2|" : "=v"(s) : "v"(a), "v"(b));
    return s;
}

__device__ __forceinline__ float tent_clamp(float s) {
    // clamp01(1 - 0.5*s) == relu(1 - 0.5*s) for s >= 0; single VALU op.
    // -0.5 and 1.0 are hardwired inline constants (no literal needed).
    float mv;
    asm("v_fma_f32_e64 %0, %1, -0.5, 1.0 clamp" : "=v"(mv) : "v"(s));
    return mv;
}

// ---------------------------------------------------------------------------
// Prep: bake per-point WMMA B-matrix lanes and fused color*alpha weights.
//   Bbuf[chunk][lane] (float4):
//     lane < 16 : ( T00, T10, T01, T11 )   -> (BtxK0, BtxK1, BtyK0, BtyK1)
//     lane >=16 : ( -c0, 0,   -c1, 0   )   -> (BtxK2, BtxK3, BtyK2, BtyK3)
//   Wbuf[n] = ( col_r*a, col_g*a, col_b*a, 0 )
// ---------------------------------------------------------------------------
__global__ void __launch_bounds__(256)
points_prep_kernel(const float* __restrict__ loc,    // (N,2)
                   const float* __restrict__ mo,     // (N,2,2)
                   const float* __restrict__ mso,    // (N,)
                   const float* __restrict__ col,    // (N,3)
                   const float* __restrict__ alp,    // (N,)
                   float4* __restrict__ Bbuf,        // [N_CHUNKS][32]
                   float4* __restrict__ Wbuf)        // [N_PTS]
{
    int n = blockIdx.x * blockDim.x + threadIdx.x;
    if (n >= N_PTS) return;

    // scale = (sqrt(N)/2) * exp(mso) = 16 * exp(mso)
    float s   = 16.0f * __expf(mso[n]);
    float T00 = mo[n * 4 + 0] + s;   // row i=0 (multiplies y)
    float T01 = mo[n * 4 + 1];
    float T10 = mo[n * 4 + 2];       // row i=1 (multiplies x)
    float T11 = mo[n * 4 + 3] + s;

    float l0 = loc[n * 2 + 0];       // pairs with y
    float l1 = loc[n * 2 + 1];       // pairs with x
    float c0 = l0 * T00 + l1 * T10;
    float c1 = l0 * T01 + l1 * T11;

    int chunk = n >> 4;
    int l     = n & 15;
    Bbuf[chunk * 32 + l]      = make_float4(T00, T10, T01, T11);
    Bbuf[chunk * 32 + 16 + l] = make_float4(-c0, 0.0f, -c1, 0.0f);

    float a = alp[n];
    Wbuf[n] = make_float4(col[n * 3 + 0] * a,
                          col[n * 3 + 1] * a,
                          col[n * 3 + 2] * a, 0.0f);
}

// ---------------------------------------------------------------------------
// Main: one wave (32 lanes) owns 16 consecutive pixels (one canvas row slice),
// loops over 64 chunks of 16 points, 2x v_wmma_f32_16x16x4_f32 per chunk.
// ---------------------------------------------------------------------------
__global__ void __launch_bounds__(256)
points_splat_kernel(const float4* __restrict__ Bbuf,
                    const float4* __restrict__ Wbuf,
                    float* __restrict__ out)
{
    const int lane    = threadIdx.x & 31;
    const int wave    = blockIdx.x * (blockDim.x >> 5) + (threadIdx.x >> 5);
    const int px_base = wave << 4;            // 16 pixels per wave

    // A matrix (16x4 f32): lanes 0-15 hold (K0,K1)=(y,x) for pixel m=lane,
    // lanes 16-31 hold (K2,K3)=(1,0). Built with selects (no EXEC divergence
    // persists into the WMMA region).
    const int   m  = lane & 15;
    const int   p  = px_base + m;             // all 16 pixels share canvas row
    const float y  = -1.0f + (2.0f / (H_CANVAS - 1)) * (float)(p >> 8);
    const float x  = -1.0f + (2.0f / (W_CANVAS - 1)) * (float)(p & (W_CANVAS - 1));
    v2f A;
    A.x = (lane < 16) ? y : 1.0f;
    A.y = (lane < 16) ? x : 0.0f;

    v8f accR = {}, accG = {}, accB = {};
    const v8f z = {};

    for (int k = 0; k < N_CHUNKS; ++k) {
        // Per-chunk parameters: coalesced b128 loads, L2-resident (48 KB total).
        float4 bb = Bbuf[k * 32 + lane];
        float4 wv = Wbuf[k * 16 + m];         // lanes L and L+16 share a point

        v2f Btx; Btx.x = bb.x; Btx.y = bb.y;
        v2f Bty; Bty.x = bb.z; Bty.y = bb.w;

        // D = A x B (+0): 16 pixels x 16 points of tx / ty, identical layouts.
        v8f dtx = __builtin_amdgcn_wmma_f32_16x16x4_f32(
            false, A, false, Btx, (short)0, z, false, false);
        v8f dty = __builtin_amdgcn_wmma_f32_16x16x4_f32(
            false, A, false, Bty, (short)0, z, false, false);

#pragma unroll
        for (int r = 0; r < 8; ++r) {
            float s  = abs_add(dtx[r], dty[r]);   // |tx|+|ty|, one VALU op
            float mv = tent_clamp(s);             // clamp01(1-0.5s), one op
            accR[r] = fmaf(wv.x, mv, accR[r]);
            accG[r] = fmaf(wv.y, mv, accG[r]);
            accB[r] = fmaf(wv.z, mv, accB[r]);
        }
    }

    // Butterfly-reduce the 16 point-columns held by each half-wave.
    // Masks 1..8 stay inside lanes [0,16) and [16,32) on wave32.
#pragma unroll
    for (int mask = 8; mask >= 1; mask >>= 1) {
#pragma unroll
        for (int r = 0; r < 8; ++r) {
            accR[r] += __shfl_xor(accR[r], mask, 32);
            accG[r] += __shfl_xor(accG[r], mask, 32);
            accB[r] += __shfl_xor(accB[r], mask, 32);
        }
    }

    // Lanes 0-15 hold rows 0-7, lanes 16-31 hold rows 8-15 (all lanes in a
    // half now identical). Lane (lane&15)==r writes canvas row r of its half.
    const int half = lane >> 4;
#pragma unroll
    for (int r = 0; r < 8; ++r) {
        if ((lane & 15) == r) {
            int prow = px_base + half * 8 + r;
            out[prow * 3 + 0] = 1.0f / (1.0f + __expf(-4.0f * accR[r]));
            out[prow * 3 + 1] = 1.0f / (1.0f + __expf(-4.0f * accG[r]));
            out[prow * 3 + 2] = 1.0f / (1.0f + __expf(-4.0f * accB[r]));
        }
    }
}

// ---------------------------------------------------------------------------
extern "C" void kernel_launch(void* const* d_in, const int* in_sizes, int n_in,
                              void* d_out, int out_size, void* d_ws, size_t ws_size,
                              hipStream_t stream)
{
    const float* locations = (const float*)d_in[0];  // (1,1,N,2)
    const float* mat_off   = (const float*)d_in[1];  // (N,2,2)
    const float* mso       = (const float*)d_in[2];  // (N,1,1)
    const float* colors    = (const float*)d_in[3];  // (1,1,N,3)
    const float* alphas    = (const float*)d_in[4];  // (1,1,N,1)
    // d_in[5], d_in[6] = canvas H, W (256, fixed by reference)

    float4* Bbuf = (float4*)d_ws;                               // 32 KB
    float4* Wbuf = (float4*)((char*)d_ws + N_CHUNKS * 32 * 16); // 16 KB
    float*  out  = (float*)d_out;

    points_prep_kernel<<<(N_PTS + 255) / 256, 256, 0, stream>>>(
        locations, mat_off, mso, colors, alphas, Bbuf, Wbuf);

    // 65536 px / 16 px-per-wave = 4096 waves; 8 waves (256 thr) per block.
    const int n_waves = (H_CANVAS * W_CANVAS) / 16;
    points_splat_kernel<<<n_waves / 8, 256, 0, stream>>>(Bbuf, Wbuf, out);
}